// _TQHEACircuit_1236950582152
// MI455X (gfx1250) — compile-verified
//
#include <hip/hip_runtime.h>
#include <math.h>

typedef __attribute__((ext_vector_type(2))) float v2f;
typedef __attribute__((ext_vector_type(8))) float v8f;

// D = A(16x4) * B(4x16) + C, f32, wave32 WMMA (CDNA5).
__device__ __forceinline__ v8f wmma4(v2f a, v2f b, v8f c) {
  return __builtin_amdgcn_wmma_f32_16x16x4_f32(
      /*neg_a=*/false, a, /*neg_b=*/false, b,
      /*c_mod=*/(short)0, c, /*reuse_a=*/false, /*reuse_b=*/false);
}

// XOR-linear LDS bank swizzle (float2 units, need distinct residues mod 32):
// low4 = b[3:0]^b[7:4]; bit4 = parity of bits {4,5,7,9,11}. Verified injective
// mod 32 for every access pattern here (gate loads/stores at sh=0/4/8, linear
// passes, ring gather) and globally bijective. swz(a^b) == swz(a)^swz(b).
__host__ __device__ constexpr int swz(int i) {
  int low4 = (i ^ (i >> 4)) & 15;
  int p = i & 0xAB0;
  p ^= p >> 8; p ^= p >> 4; p ^= p >> 2; p ^= p >> 1;
  return (i & 0xFE0) | ((p & 1) << 4) | low4;
}

// CNOT-ring (sequential CNOT(i,i+1 mod 12)) inverse permutation, XOR-linear:
// source index feeding destination d.
__host__ __device__ constexpr int srcmap(int d) {
  int s = (d ^ (d >> 1)) & 0x3FF;
  s |= (((d >> 10) ^ (d >> 11) ^ d) & 1) << 10;
  s |= (((d >> 11) ^ d) & 1) << 11;
  return s;
}

// A-matrix LDS row relabel: rows k with bit1 set swap with k^1 so the paired
// float2 b64 loads (rows kA,kA+1; half-waves kA vs kA+2) are conflict-free.
__host__ __device__ constexpr int aslot(int t) {
  return t ^ (((t >> 5) & 1) << 4);
}

// ---------------------------------------------------------------------------
// Prologue: fused per-layer 4-wire unitaries U = kron_j RY(a2)RZ(a1)RY(a0),
// 6 layers x 3 quads, TRANSPOSED + interleaved: wsU[(L*3+q)*256 + k*16+m] =
// {Re,Im} of U[m][k]. Batch independent.
// ---------------------------------------------------------------------------
__global__ void build_ansatz_u(const float* __restrict__ W, float2* __restrict__ wsU) {
  int tid = blockIdx.x * 256 + threadIdx.x;
  if (tid >= 6 * 3 * 256) return;
  int L = tid / 768;
  int q = (tid / 256) % 3;
  int k = (tid >> 4) & 15;
  int m = tid & 15;

  float pr = 1.f, pi = 0.f;
#pragma unroll
  for (int j = 0; j < 4; ++j) {
    int w = 4 * q + j;                 // wire (quad-internal bit 3-j)
    float a0 = W[L * 36 + 0 * 12 + w];
    float a1 = W[L * 36 + 1 * 12 + w];
    float a2 = W[L * 36 + 2 * 12 + w];
    float c0 = cosf(a0 * 0.5f), s0 = sinf(a0 * 0.5f);
    float ct = cosf(a1 * 0.5f), st = sinf(a1 * 0.5f);
    float c2 = cosf(a2 * 0.5f), s2 = sinf(a2 * 0.5f);
    int mb = (m >> (3 - j)) & 1, kb = (k >> (3 - j)) & 1;
    // U_w = A*e^{-i t1} + B*e^{+i t1}
    float A, B;
    if (!mb && !kb)      { A =  c2 * c0; B = -s2 * s0; }
    else if (!mb &&  kb) { A = -c2 * s0; B = -s2 * c0; }
    else if ( mb && !kb) { A =  s2 * c0; B =  c2 * s0; }
    else                 { A = -s2 * s0; B =  c2 * c0; }
    float ur = (A + B) * ct;
    float ui = (B - A) * st;
    float nr = pr * ur - pi * ui;
    float ni = pr * ui + pi * ur;
    pr = nr; pi = ni;
  }
  wsU[(L * 3 + q) * 256 + k * 16 + m] = make_float2(pr, pi);
}

// ---------------------------------------------------------------------------
// Apply staged 16x16 complex unitary (At, transposed [k][m], rows relabelled
// by aslot) to the 4-bit index field of quad Q. Each wave owns 2 disjoint
// 16-column tiles -> no barrier inside. All swz(K<<sh) terms are constants.
// ---------------------------------------------------------------------------
template <int Q>
__device__ __forceinline__ void apply_gate(float2* amp, const float2* At,
                                           int nn, int kh, int wv) {
  constexpr int sh = 8 - 4 * Q;
  constexpr int lowmask = (1 << sh) - 1;
  const int skb = (0 - kh) & swz(2 << sh);   // swz((2*kh)<<sh)
  const int smh = (0 - kh) & swz(8 << sh);   // swz((8*kh)<<sh)
  const int a0b = 48 * kh + nn;              // relabelled row kA,   col nn
  const int a1b = 16 + 16 * kh + nn;         // relabelled row kA+1, col nn
#pragma unroll
  for (int t = 0; t < 2; ++t) {
    const int g = (wv * 2 + t) * 16 + nn;                     // column N
    const int base = ((g >> sh) << (sh + 4)) | (g & lowmask); // field bits = 0
    const int sb = swz(base);
    const int sbl = sb ^ skb;
    const int sbs = sb ^ smh;
    v8f accP = {0.f,0.f,0.f,0.f,0.f,0.f,0.f,0.f};
    v8f accQ = {0.f,0.f,0.f,0.f,0.f,0.f,0.f,0.f};
    v8f accI = {0.f,0.f,0.f,0.f,0.f,0.f,0.f,0.f};
#pragma unroll
    for (int kk = 0; kk < 4; ++kk) {
      float2 A0 = At[64 * kk + a0b];         // (Re,Im) of U[nn][kA]
      float2 A1 = At[64 * kk + a1b];         // (Re,Im) of U[nn][kA+1]
      v2f a_r = {A0.x, A1.x};
      v2f a_i = {A0.y, A1.y};
      float2 b0 = amp[sbl ^ swz((4 * kk) << sh)];      // K=kA,   N=nn
      float2 b1 = amp[sbl ^ swz((4 * kk + 1) << sh)];  // K=kA+1, N=nn
      v2f b_r = {b0.x, b1.x};
      v2f b_i = {b0.y, b1.y};
      accP = wmma4(a_r, b_r, accP);   // Ur*Vr
      accQ = wmma4(a_i, b_i, accQ);   // Ui*Vi
      accI = wmma4(a_r, b_i, accI);   // Ur*Vi + Ui*Vr
      accI = wmma4(a_i, b_r, accI);
    }
#pragma unroll
    for (int e = 0; e < 8; ++e) {     // D: M=e+8*kh, N=nn
      amp[sbs ^ swz(e << sh)] = make_float2(accP[e] - accQ[e], accI[e]);
    }
  }
}

// ---------------------------------------------------------------------------
// Main: one workgroup (256 thr = 8 waves) per batch element, state in LDS.
// ---------------------------------------------------------------------------
__global__ __launch_bounds__(256) void tqhea_kernel(const float* __restrict__ x,
                                                    const float2* __restrict__ wsU,
                                                    float* __restrict__ out) {
  __shared__ float2 amp[4096];
  __shared__ __align__(8) float2 At[256];
  __shared__ float accum;

  const int tid = threadIdx.x;
  const int b = blockIdx.x;
  const int lane = tid & 31;
  const int wv = tid >> 5;
  const int nn = lane & 15;
  const int kh = lane >> 4;
  const int stid = swz(tid);           // for linear passes
  const int ptid = swz(srcmap(tid));   // for ring gather
  const float* xb = x + b * 24;

  if (tid == 0) accum = 0.f;
#pragma unroll
  for (int i = 0; i < 16; ++i) {
    int idx = i * 256 + tid;           // bijection over slots; swz(0)==0
    amp[idx] = (idx == 0) ? make_float2(1.f, 0.f) : make_float2(0.f, 0.f);
  }
  __syncthreads();

  for (int half = 0; half < 2; ++half) {
    // ---- encode: 3 fused RX quads; entry depends only on m^k (symmetric).
    auto stage_rx = [&](int q) {
      int k = tid >> 4, m = tid & 15;
      int xv = m ^ k;
      float r = 1.f;
#pragma unroll
      for (int j = 0; j < 4; ++j) {
        float th = xb[half * 12 + 4 * q + j] * 0.5f;
        r *= ((xv >> (3 - j)) & 1) ? sinf(th) : cosf(th);
      }
      int d4 = __popc(xv) & 3;         // (-i)^d4
      float rr = (d4 == 0) ? r : ((d4 == 2) ? -r : 0.f);
      float ri = (d4 == 1) ? -r : ((d4 == 3) ? r : 0.f);
      At[aslot(tid)] = make_float2(rr, ri);
    };
    stage_rx(0); __syncthreads(); apply_gate<0>(amp, At, nn, kh, wv); __syncthreads();
    stage_rx(1); __syncthreads(); apply_gate<1>(amp, At, nn, kh, wv); __syncthreads();
    stage_rx(2); __syncthreads(); apply_gate<2>(amp, At, nn, kh, wv); __syncthreads();

    // ---- 3 ansatz layers
    for (int l = 0; l < 3; ++l) {
      int L = half * 3 + l;
      auto stage_u = [&](int q) { At[aslot(tid)] = wsU[(L * 3 + q) * 256 + tid]; };
      stage_u(0); __syncthreads(); apply_gate<0>(amp, At, nn, kh, wv); __syncthreads();
      stage_u(1); __syncthreads(); apply_gate<1>(amp, At, nn, kh, wv); __syncthreads();
      stage_u(2); __syncthreads(); apply_gate<2>(amp, At, nn, kh, wv); __syncthreads();

      // ---- CNOT ring = one GF(2) bit permutation; addresses strength-
      // reduced via linearity: swz(srcmap(i*256+tid)) = const_i ^ ptid.
      float2 tmp[16];
#pragma unroll
      for (int i = 0; i < 16; ++i) tmp[i] = amp[swz(srcmap(i * 256)) ^ ptid];
      __syncthreads();
#pragma unroll
      for (int i = 0; i < 16; ++i) amp[swz(i * 256) ^ stid] = tmp[i];
      __syncthreads();
    }
  }

  // ---- <H> = sum |a_k|^2 * (12 - 2*popcount(k)); popc splits over i|tid.
  const int pt = __popc(tid);
  float e = 0.f;
#pragma unroll
  for (int i = 0; i < 16; ++i) {
    float2 a = amp[swz(i * 256) ^ stid];
    e += (a.x * a.x + a.y * a.y) * (float)(12 - 2 * (__popc(i) + pt));
  }
#pragma unroll
  for (int off = 16; off > 0; off >>= 1) e += __shfl_down(e, off, 32);
  if (lane == 0) atomicAdd(&accum, e);
  __syncthreads();
  if (tid == 0) out[b] = accum;
}

extern "C" void kernel_launch(void* const* d_in, const int* in_sizes, int n_in,
                              void* d_out, int out_size, void* d_ws, size_t ws_size,
                              hipStream_t stream) {
  (void)n_in; (void)ws_size; (void)out_size;
  const float* x = (const float*)d_in[0];          // (B, 24) f32
  const float* w = (const float*)d_in[1];          // (6, 3, 12) f32
  float* out = (float*)d_out;                      // (B, 1) f32
  float2* wsU = (float2*)d_ws;                     // 4608 float2 fused unitaries

  int B = in_sizes[0] / 24;

  build_ansatz_u<<<18, 256, 0, stream>>>(w, wsU);
  tqhea_kernel<<<B, 256, 0, stream>>>(x, wsU, out);
}